// Octave_4544075399626
// MI455X (gfx1250) — compile-verified
//
#include <hip/hip_runtime.h>

typedef __attribute__((ext_vector_type(2))) float v2f;
typedef __attribute__((ext_vector_type(8))) float v8f;

namespace {
constexpr int   kN       = 512;             // output grid per cube face
constexpr int   kM       = 256;             // input grid per cube face
constexpr float kScale   = 255.0f / 511.0f; // align-corners step (m-1)/(n-1)
constexpr float kFactor  = 0.5f;
constexpr float kEps     = 1e-12f;
constexpr int   kFacePix = kN * kN;         // 262144
constexpr int   kVerts   = 6 * kFacePix;    // 1572864
constexpr int   kTiles   = kN / 16;         // 32 tiles per axis
constexpr int   kKr      = 12;              // padded K extent per 16-row output tile
}

// ---------------------------------------------------------------------------
// Kernel A: fused bilinear resize (align-corners) + sigmoid displacement.
// One wave == one 16x16 output tile of one cube face. The x-axis resample is a
// 2-tap gather into LDS; the y-axis resample is a true matmul
// D(16x16) = A(16x12) * B(12x16) done as 3x V_WMMA_F32_16X16X4_F32 per channel.
// All 6 channels (vrt.xyz, nrm.xyz) share the A matrix (interp weights).
// ---------------------------------------------------------------------------
__global__ __launch_bounds__(32)
void resize_displace_wmma(const float* __restrict__ vrt,
                          const float* __restrict__ nrm,
                          const float* __restrict__ ps,
                          float* __restrict__ vert_out)
{
    __shared__ float sB[6][kKr][16];

    const int lane = threadIdx.x;                 // 0..31, exactly one wave32
    const int t    = blockIdx.x;
    const int f    = t / (kTiles * kTiles);
    const int rem  = t - f * (kTiles * kTiles);
    const int ty   = rem / kTiles;
    const int tx   = rem - ty * kTiles;

    const int r0 = (int)floorf((float)(ty * 16) * kScale);  // first src row used

    // ---- stage 1: x-axis 2-tap resample of the 12-row input strip into LDS.
    // 6 channels x 12 rows x 16 cols = 1152 elements, 36 per lane.
    for (int e = lane; e < 6 * kKr * 16; e += 32) {
        const int ch = e / (kKr * 16);
        const int k  = (e - ch * (kKr * 16)) / 16;
        const int j  = e & 15;
        int rk = r0 + k;
        rk = rk < 0 ? 0 : (rk > kM - 1 ? kM - 1 : rk);
        const float x  = (float)(tx * 16 + j) * kScale;
        const int   j0 = (int)floorf(x);
        const float wx = x - (float)j0;
        const int   j1 = (j0 + 1 > kM - 1) ? kM - 1 : j0 + 1;
        const float* src = (ch < 3) ? vrt : nrm;
        const int    c   = (ch < 3) ? ch : ch - 3;
        const int base   = (f * kM * kM + rk * kM) * 3 + c;
        const float lo = src[base + j0 * 3];
        const float hi = src[base + j1 * 3];
        sB[ch][k][j] = lo * (1.0f - wx) + hi * wx;
    }
    __syncthreads();

    // ---- stage 2: y-axis resample via WMMA.
    // A (32-bit, 16x4) lane layout: lanes 0-15 hold rows M=0..15 with K=0 (v0)
    // and K=1 (v1); lanes 16-31 hold K=2/K=3.
    const int   aM   = lane & 15;
    const int   kHi  = (lane >> 4) << 1;          // 0 or 2
    const int   rOut = ty * 16 + aM;
    const float y    = (float)rOut * kScale;
    const int   i0   = (int)floorf(y);
    const float wy   = y - (float)i0;
    const int   i1   = (i0 + 1 > kM - 1) ? kM - 1 : i0 + 1;

    v2f aChunk[3];
    #pragma unroll
    for (int kc = 0; kc < 3; ++kc) {
        #pragma unroll
        for (int q = 0; q < 2; ++q) {
            const int kg = r0 + kc * 4 + kHi + q;
            float a = 0.0f;
            if (kg == i0) a += 1.0f - wy;         // handles i0==i1 at the edge
            if (kg == i1) a += wy;
            aChunk[kc][q] = a;
        }
    }

    // B (32-bit, 4x16) lane layout: lanes 0-15 hold cols N=0..15 with K=0/K=1,
    // lanes 16-31 hold K=2/K=3 (mirror of the A layout).
    const int bN  = lane & 15;
    const int bHi = (lane >> 4) << 1;

    v8f acc[6];
    #pragma unroll
    for (int ch = 0; ch < 6; ++ch) {
        v8f c = {};
        #pragma unroll
        for (int kc = 0; kc < 3; ++kc) {
            v2f b;
            b[0] = sB[ch][kc * 4 + bHi + 0][bN];
            b[1] = sB[ch][kc * 4 + bHi + 1][bN];
            // (neg_a, A, neg_b, B, c_mod, C, reuse_a, reuse_b)
            c = __builtin_amdgcn_wmma_f32_16x16x4_f32(false, aChunk[kc],
                                                      false, b,
                                                      (short)0, c,
                                                      false, false);
        }
        acc[ch] = c;
    }

    // ---- stage 3: displacement + store.
    // C/D layout: VGPR g holds row M=g (lanes 0-15) or M=g+8 (lanes 16-31),
    // N = lane % 16.
    const int cCol = tx * 16 + (lane & 15);
    const int rAdd = (lane >> 4) << 3;
    #pragma unroll
    for (int g = 0; g < 8; ++g) {
        const int r   = ty * 16 + g + rAdd;
        const int pix = f * kFacePix + r * kN + cCol;
        const float nx = acc[3][g], ny = acc[4][g], nz = acc[5][g];
        const float mag = sqrtf(nx * nx + ny * ny + nz * nz);
        const float sig = 1.0f / (1.0f + expf(-ps[pix]));
        const float d   = kFactor * sig * mag;
        const int o = pix * 3;
        vert_out[o + 0] = acc[0][g] + d;
        vert_out[o + 1] = acc[1][g] + d;
        vert_out[o + 2] = acc[2][g] + d;
    }
}

// ---------------------------------------------------------------------------
// Kernel B: vertex normals. Deterministic gather (NO float atomics): each
// vertex recomputes its <=6 incident triangle normals in a fixed order. The
// 19 MB vert buffer is fully resident in the 192 MB L2, so the redundant
// reads are cheap; output traffic dominates.
// ---------------------------------------------------------------------------
struct F3 { float x, y, z; };

__device__ __forceinline__ F3 ldv(const float* __restrict__ v, int f, int i, int j) {
    const int o = (f * kFacePix + i * kN + j) * 3;
    return F3{v[o], v[o + 1], v[o + 2]};
}

__device__ __forceinline__ F3 triN(F3 a, F3 b, F3 c) {
    const float e1x = b.x - a.x, e1y = b.y - a.y, e1z = b.z - a.z;
    const float e2x = c.x - a.x, e2y = c.y - a.y, e2z = c.z - a.z;
    const float fx = e1y * e2z - e1z * e2y;
    const float fy = e1z * e2x - e1x * e2z;
    const float fz = e1x * e2y - e1y * e2x;
    const float n   = sqrtf(fx * fx + fy * fy + fz * fz);
    const float inv = 1.0f / fmaxf(n, kEps);
    return F3{fx * inv, fy * inv, fz * inv};
}

__global__ __launch_bounds__(256)
void vertex_normals(const float* __restrict__ vert, float* __restrict__ nout)
{
    const int v = blockIdx.x * blockDim.x + threadIdx.x;
    if (v >= kVerts) return;
    const int f   = v >> 18;                 // / 262144
    const int rem = v & (kFacePix - 1);
    const int i   = rem >> 9;                // row in 512x512 grid
    const int j   = rem & (kN - 1);          // col

    // Warm L2/WGP$ for the 3-row neighborhood (lowers to global_prefetch_b8).
    __builtin_prefetch(&vert[(f * kFacePix + i * kN + j) * 3], 0, 0);

    float ax = 0.0f, ay = 0.0f, az = 0.0f;
    #define ACC3(p0, p1, p2) do { F3 _n = triN(p0, p1, p2); \
        ax += _n.x; ay += _n.y; az += _n.z; } while (0)

    const bool iL = (i >= 1), iH = (i < kN - 1);
    const bool jL = (j >= 1), jH = (j < kN - 1);

    // Triangulation: cell(r,c) -> tri1 {(r,c),(r,c+1),(r+1,c)},
    //                             tri2 {(r,c+1),(r+1,c+1),(r+1,c)}.
    if (iH && jH) ACC3(ldv(vert,f,i,  j  ), ldv(vert,f,i,  j+1), ldv(vert,f,i+1,j  )); // tri1(i,j)
    if (iH && jL) ACC3(ldv(vert,f,i,  j-1), ldv(vert,f,i,  j  ), ldv(vert,f,i+1,j-1)); // tri1(i,j-1)
    if (iL && jH) ACC3(ldv(vert,f,i-1,j  ), ldv(vert,f,i-1,j+1), ldv(vert,f,i,  j  )); // tri1(i-1,j)
    if (iH && jL) ACC3(ldv(vert,f,i,  j  ), ldv(vert,f,i+1,j  ), ldv(vert,f,i+1,j-1)); // tri2(i,j-1)
    if (iL && jL) ACC3(ldv(vert,f,i-1,j  ), ldv(vert,f,i,  j  ), ldv(vert,f,i,  j-1)); // tri2(i-1,j-1)
    if (iL && jH) ACC3(ldv(vert,f,i-1,j+1), ldv(vert,f,i,  j+1), ldv(vert,f,i,  j  )); // tri2(i-1,j)
    #undef ACC3

    const float len = sqrtf(ax * ax + ay * ay + az * az);
    const float inv = 1.0f / fmaxf(len, kEps);
    const float sgn[6] = {-1.0f, 1.0f, 1.0f, -1.0f, -1.0f, 1.0f};
    const float s = sgn[f] * inv;
    const int o = v * 3;
    nout[o + 0] = ax * s;
    nout[o + 1] = ay * s;
    nout[o + 2] = az * s;
}

// ---------------------------------------------------------------------------
// Host launcher. Output layout (flat, reference return order):
//   [0, 3V)                     vert  (f32)
//   [3V, 3V + |faces|)          faces (int32 bit-pattern, straight d2d copy)
//   [3V + |faces|, ... + 3V)    normals (f32)
// ---------------------------------------------------------------------------
extern "C" void kernel_launch(void* const* d_in, const int* in_sizes, int n_in,
                              void* d_out, int out_size, void* d_ws, size_t ws_size,
                              hipStream_t stream)
{
    const float* vrt   = (const float*)d_in[0];
    const float* nrm   = (const float*)d_in[1];
    const float* ps    = (const float*)d_in[2];
    const int*   faces = (const int*)d_in[3];

    float* out       = (float*)d_out;
    float* vert_out  = out;
    const size_t vertFloats = (size_t)kVerts * 3;
    float* faces_out = out + vertFloats;
    float* norm_out  = faces_out + (size_t)in_sizes[3];

    // 1) resize + displacement -> vert (WMMA path)
    resize_displace_wmma<<<dim3(6 * kTiles * kTiles), dim3(32), 0, stream>>>(
        vrt, nrm, ps, vert_out);

    // 2) faces passthrough (same 4-byte width, bitwise copy)
    hipMemcpyAsync(faces_out, faces, (size_t)in_sizes[3] * sizeof(int),
                   hipMemcpyDeviceToDevice, stream);

    // 3) vertex normals from the freshly written vert buffer (stream-ordered)
    vertex_normals<<<dim3((kVerts + 255) / 256), dim3(256), 0, stream>>>(
        vert_out, norm_out);

    (void)n_in; (void)out_size; (void)d_ws; (void)ws_size;
}